// DDAU_Encoder_12841952215142
// MI455X (gfx1250) — compile-verified
//
#include <hip/hip_runtime.h>
#include <hip/hip_bf16.h>

#define USER_NUM 100000
#define ITEM_NUM 50000
#define NNODES   (USER_NUM + ITEM_NUM)   // 150000
#define EMB      64
#define N_LAYERS 3
#define EPS_F    0.1f

typedef __attribute__((ext_vector_type(2))) float v2f;
typedef __attribute__((ext_vector_type(8))) float v8f;

__device__ __forceinline__ float sgnf(float x) {
    return (x > 0.0f) ? 1.0f : ((x < 0.0f) ? -1.0f : 0.0f);
}

// ---------------------------------------------------------------------------
// ego <- [user_emb ; item_emb] ; acc <- 0   (float4 grid-stride)
// ---------------------------------------------------------------------------
__global__ void init_kernel(const float* __restrict__ ue, const float* __restrict__ ie,
                            float* __restrict__ ego, float* __restrict__ acc) {
    const long long total4 = (long long)NNODES * EMB / 4;
    const long long ub4    = (long long)USER_NUM * EMB / 4;
    long long t      = (long long)blockIdx.x * blockDim.x + threadIdx.x;
    long long stride = (long long)gridDim.x * blockDim.x;
    const float4 z = make_float4(0.f, 0.f, 0.f, 0.f);
    for (long long i = t; i < total4; i += stride) {
        float4 v = (i < ub4) ? ((const float4*)ue)[i] : ((const float4*)ie)[i - ub4];
        ((float4*)ego)[i] = v;
        ((float4*)acc)[i] = z;
    }
}

// ---------------------------------------------------------------------------
// CSR build step 1: zero per-row counts
// ---------------------------------------------------------------------------
__global__ void zero_cnt_kernel(int* __restrict__ cnt) {
    int t = blockIdx.x * blockDim.x + threadIdx.x;
    int stride = gridDim.x * blockDim.x;
    for (int i = t; i < NNODES; i += stride) cnt[i] = 0;
}

// CSR build step 2: histogram of destination rows (int atomics -> exact)
__global__ void count_kernel(const int* __restrict__ rows, int* __restrict__ cnt,
                             long long n_edges) {
    long long t = (long long)blockIdx.x * blockDim.x + threadIdx.x;
    long long stride = (long long)gridDim.x * blockDim.x;
    for (long long e = t; e < n_edges; e += stride)
        atomicAdd(&cnt[rows[e]], 1);
}

// ---------------------------------------------------------------------------
// CSR build step 3: exclusive scan of counts -> row_start[N+1]; seed cursors.
// Single block, 1024 threads, chunked serial + Hillis-Steele over thread sums.
// Fully deterministic.
// ---------------------------------------------------------------------------
__global__ __launch_bounds__(1024)
void scan_kernel(const int* __restrict__ cnt, int* __restrict__ row_start,
                 int* __restrict__ cursor) {
    __shared__ int lds[1024];
    const int t = threadIdx.x;
    const int chunk = (NNODES + 1023) / 1024;        // 147
    const int lo = t * chunk;
    const int hi = min(NNODES, lo + chunk);
    int s = 0;
    for (int i = lo; i < hi; ++i) s += cnt[i];
    lds[t] = s;
    __syncthreads();
    for (int off = 1; off < 1024; off <<= 1) {
        int v = (t >= off) ? lds[t - off] : 0;
        __syncthreads();
        lds[t] += v;
        __syncthreads();
    }
    int run = (t == 0) ? 0 : lds[t - 1];
    for (int i = lo; i < hi; ++i) {
        row_start[i] = run;
        cursor[i]    = run;
        run += cnt[i];
    }
    if (lo < NNODES && hi == NNODES) row_start[NNODES] = run;
}

// CSR build step 4: scatter edges into CSR arrays
__global__ void scatter_kernel(const int* __restrict__ rows, const int* __restrict__ cols,
                               const float* __restrict__ vals, int* __restrict__ cursor,
                               int* __restrict__ csr_col, float* __restrict__ csr_val,
                               long long n_edges) {
    long long t = (long long)blockIdx.x * blockDim.x + threadIdx.x;
    long long stride = (long long)gridDim.x * blockDim.x;
    for (long long e = t; e < n_edges; e += stride) {
        int p = atomicAdd(&cursor[rows[e]], 1);
        csr_col[p] = cols[e];
        csr_val[p] = vals[e];
    }
}

// ---------------------------------------------------------------------------
// Pull-based SpMM fused with perturbation + accumulation. One wave32 per row:
//   y[row] = sum_e val_e * x[col_e]            (no atomics; gathers hit L2)
//   e      = y + sign(y) * (noise/||noise||) * EPS
//   ego_out[row] = e ; acc[row] += e
// Lanes load 32 edges coalesced, broadcast col/val via shuffle, each gather is
// one coalesced 256B global_load_b64 across the wave.
// ---------------------------------------------------------------------------
__global__ void pull_spmm_update_kernel(const int* __restrict__ row_start,
                                        const int* __restrict__ csr_col,
                                        const float* __restrict__ csr_val,
                                        const float* __restrict__ x,
                                        const float* __restrict__ noise_k,
                                        float* __restrict__ ego_out,
                                        float* __restrict__ acc) {
    int wave   = (blockIdx.x * blockDim.x + threadIdx.x) >> 5;
    int lane   = threadIdx.x & 31;
    int nwaves = (gridDim.x * blockDim.x) >> 5;
    for (int row = wave; row < NNODES; row += nwaves) {
        const int s = row_start[row];
        const int e = row_start[row + 1];
        float2 y = make_float2(0.f, 0.f);
        for (int j = s; j < e; j += 32) {
            int idx = j + lane;
            int   cc = 0;
            float vv = 0.f;
            if (idx < e) {
                cc = csr_col[idx];
                vv = csr_val[idx];
            }
            if (j + 64 < e) {                       // prefetch next edge chunk
                __builtin_prefetch(csr_col + j + 32 + lane, 0, 0);
                __builtin_prefetch(csr_val + j + 32 + lane, 0, 0);
            }
            const int nv = min(32, e - j);          // wave-uniform
            for (int k = 0; k < nv; ++k) {
                int   c2 = __shfl(cc, k, 32);
                float v2 = __shfl(vv, k, 32);
                float2 xv = *(const float2*)(x + (long long)c2 * EMB + lane * 2);
                y.x = fmaf(v2, xv.x, y.x);
                y.y = fmaf(v2, xv.y, y.y);
            }
        }
        // fused noise perturbation + accumulate
        long long base = (long long)row * EMB + lane * 2;
        float2 nz = *(const float2*)(noise_k + base);
        float ss = nz.x * nz.x + nz.y * nz.y;
        #pragma unroll
        for (int off = 16; off > 0; off >>= 1) ss += __shfl_xor(ss, off, 32);
        float inv = EPS_F / fmaxf(sqrtf(ss), 1e-12f);
        float2 ev;
        ev.x = y.x + sgnf(y.x) * nz.x * inv;
        ev.y = y.y + sgnf(y.y) * nz.y * inv;
        *(float2*)(ego_out + base) = ev;
        float2 a = *(float2*)(acc + base);
        a.x += ev.x; a.y += ev.y;
        *(float2*)(acc + base) = a;
    }
}

// ---------------------------------------------------------------------------
// Epilogue: out = acc * (1/3), one 16x16 tile per wave via
// D = A x (1/3 * I16) as 4 chained V_WMMA_F32_16X16X4_F32 (K=16 in 4 chunks).
// ---------------------------------------------------------------------------
__global__ void scale_wmma_kernel(const float* __restrict__ acc, float* __restrict__ out) {
    const float s = 1.0f / (float)N_LAYERS;
    int wave   = (blockIdx.x * blockDim.x + threadIdx.x) >> 5;
    int nwaves = (gridDim.x * blockDim.x) >> 5;
    int lane   = threadIdx.x & 31;
    int half   = lane >> 4;       // 0: lanes 0-15, 1: lanes 16-31
    int l16    = lane & 15;
    const int tiles = (NNODES / 16) * (EMB / 16);   // 9375 * 4 = 37500
    for (int tile = wave; tile < tiles; tile += nwaves) {
        int rb = (tile >> 2) * 16;       // node-row base
        int cb = (tile & 3) * 16;        // emb-column base
        v8f c = {};
        #pragma unroll
        for (int g = 0; g < 4; ++g) {    // K chunks of 4
            int k0 = half * 2;           // K index held in VGPR0 on this lane half
            const float* arow = acc + (long long)(rb + l16) * EMB + cb + g * 4;
            v2f a; a.x = arow[k0]; a.y = arow[k0 + 1];
            v2f b;                       // (1/3)*I chunk: B[k][n] = s * delta(4g+k, n)
            b.x = (g * 4 + k0     == l16) ? s : 0.0f;
            b.y = (g * 4 + k0 + 1 == l16) ? s : 0.0f;
            c = __builtin_amdgcn_wmma_f32_16x16x4_f32(false, a, false, b,
                                                      (short)0, c, false, false);
        }
        #pragma unroll
        for (int r = 0; r < 8; ++r) {
            int m = r + 8 * half;
            out[(long long)(rb + m) * EMB + cb + l16] = c[r];
        }
    }
}

// ---------------------------------------------------------------------------
extern "C" void kernel_launch(void* const* d_in, const int* in_sizes, int n_in,
                              void* d_out, int out_size, void* d_ws, size_t ws_size,
                              hipStream_t stream) {
    const float* user_emb = (const float*)d_in[0];
    const float* item_emb = (const float*)d_in[1];
    const float* u_proto  = (const float*)d_in[2];
    const float* i_proto  = (const float*)d_in[3];
    const int*   rows     = (const int*)d_in[4];
    const int*   cols     = (const int*)d_in[5];
    const float* vals     = (const float*)d_in[6];
    const float* noise    = (const float*)d_in[7];
    const long long n_edges = (long long)in_sizes[4];

    float* out = (float*)d_out;

    // workspace layout
    const size_t NE = (size_t)NNODES * EMB;          // 9.6M floats
    float* ego_a    = (float*)d_ws;                  // 38.4 MB
    float* ego_b    = ego_a + NE;                    // 38.4 MB
    float* acc      = ego_b + NE;                    // 38.4 MB
    float* csr_val  = acc + NE;                      // 32 MB
    int*   csr_col  = (int*)(csr_val + (size_t)n_edges);   // 32 MB
    int*   row_start= csr_col + (size_t)n_edges;     // (N+1) ints
    int*   cursor   = row_start + (NNODES + 1);      // N ints
    int*   cnt      = cursor + NNODES;               // N ints

    // ---- CSR build (once per call) ----
    zero_cnt_kernel<<<586, 256, 0, stream>>>(cnt);
    count_kernel<<<8192, 256, 0, stream>>>(rows, cnt, n_edges);
    scan_kernel<<<1, 1024, 0, stream>>>(cnt, row_start, cursor);
    scatter_kernel<<<8192, 256, 0, stream>>>(rows, cols, vals, cursor,
                                             csr_col, csr_val, n_edges);

    // ---- init embeddings / accumulator ----
    init_kernel<<<9376, 256, 0, stream>>>(user_emb, item_emb, ego_a, acc);

    // ---- 3 propagation layers (ping-pong, fused update) ----
    float* cur = ego_a;
    float* nxt = ego_b;
    for (int k = 0; k < N_LAYERS; ++k) {
        pull_spmm_update_kernel<<<18750, 256, 0, stream>>>(
            row_start, csr_col, csr_val, cur, noise + (size_t)k * NE, nxt, acc);
        float* tmp = cur; cur = nxt; nxt = tmp;
    }

    // ---- out[0 : 150000*64] = acc / 3 via WMMA ----
    scale_wmma_kernel<<<4096, 256, 0, stream>>>(acc, out);

    // ---- prototype pass-through ----
    const size_t proto_elems = (size_t)in_sizes[2];   // 4000*64
    hipMemcpyAsync(out + NE, u_proto, proto_elems * sizeof(float),
                   hipMemcpyDeviceToDevice, stream);
    hipMemcpyAsync(out + NE + proto_elems, i_proto, (size_t)in_sizes[3] * sizeof(float),
                   hipMemcpyDeviceToDevice, stream);
}